// FNNKernelTransform_2551210574747
// MI455X (gfx1250) — compile-verified
//
#include <hip/hip_runtime.h>

typedef float v2f __attribute__((ext_vector_type(2)));
typedef float v4f __attribute__((ext_vector_type(4)));
typedef float v8f __attribute__((ext_vector_type(8)));

#define HID    128        // hidden width
#define HP     132        // padded h-dim: 128 weights + 1 bias row + 3 zeros (mult of 4)
#define NJ     33         // v_aug length (32 + bias col)
#define VP     36         // padded v stride (16B-aligned rows)
#define KDIM   (NJ * HP)  // 4356 contraction length of the big GEMM
#define KSTEPS (KDIM / 4) // 1089 WMMA k-steps
#define JBLK   33         // k-steps per j value (HP/4)
#define CHUNK  132        // k-steps per LDS chunk = 4 j-blocks (67.5 KB)
#define NCHUNK 9          // 8 full chunks (4 j each) + 1 tail chunk (1 j)
#define EPW    32         // edges per wave (2 M-tiles)
#define WAVES  4
#define EPB    (EPW * WAVES) // 128 edges per block

// d_ws layout (floats): weights prepacked into exact B-fragment order so both
// the global->LDS stage and the LDS->VGPR fragment read are single b64/b128 ops.
#define W1F_OFF 0
#define W1F_SZ  512                 // 1 kstep * 8 ntiles * 64
#define W2F_OFF (W1F_OFF + W1F_SZ)
#define W2F_SZ  16384               // 32 ksteps * 8 ntiles * 64
#define WFF_OFF (W2F_OFF + W2F_SZ)
#define WFF_SZ  (KSTEPS * 128)      // 1089 ksteps * 2 ntiles * 64
#define WS_FLOATS (WFF_OFF + WFF_SZ)

// ---------------------------------------------------------------------------
// Prepack: gather W1/W2/W3+b3 into WMMA B-fragment order.
// Fragment layout convention (f32 16x16x4): lane l supplies N = l&15,
// K = p + 2*(l>>4) for VGPR p in {0,1}.  Big-GEMM k index: k = j*132 + h,
// Wf[k, i] = (h<128 ? W3[h, i*33+j] : (h==128 ? b3[i*33+j] : 0)).
// ---------------------------------------------------------------------------
__global__ __launch_bounds__(256) void prepack_kernel(
    const float* __restrict__ W1, const float* __restrict__ W2,
    const float* __restrict__ W3, const float* __restrict__ b3,
    float* __restrict__ ws)
{
  int idx = blockIdx.x * blockDim.x + threadIdx.x;
  if (idx >= WS_FLOATS) return;
  if (idx < W2F_OFF) {                       // W1 fragments (4 x 128)
    int t = idx - W1F_OFF;
    int p = t & 1, l = (t >> 1) & 31, n = t >> 6;
    int K = p + 2 * (l >> 4);
    int col = n * 16 + (l & 15);
    ws[idx] = W1[K * HID + col];
  } else if (idx < WFF_OFF) {                // W2 fragments (128 x 128)
    int t = idx - W2F_OFF;
    int p = t & 1, l = (t >> 1) & 31, tn = t >> 6;
    int n = tn & 7, kk = tn >> 3;
    int K = 4 * kk + p + 2 * (l >> 4);
    int col = n * 16 + (l & 15);
    ws[idx] = W2[K * HID + col];
  } else {                                   // Wf fragments (4356 x 32)
    int t = idx - WFF_OFF;
    int p = t & 1, l = (t >> 1) & 31, sn = t >> 6;
    int n = sn & 1, s = sn >> 1;
    int kidx = 4 * s + p + 2 * (l >> 4);
    int j = kidx / HP;
    int h = kidx - j * HP;
    int col = (n * 16 + (l & 15)) * NJ + j;  // i*33 + j
    float v = 0.0f;
    if (h < HID)       v = W3[h * 1056 + col];
    else if (h == HID) v = b3[col];
    ws[idx] = v;
  }
}

// ---------------------------------------------------------------------------
// Fused kernel: layer1+layer2 via WMMA f32 16x16x4, then the big T-GEMM
// out(E x 32) = T(E x 4356) @ Wf(4356 x 32) with T built per k-step as a
// rank-1 product h2_aug[e,h] * v_aug[e,j] from LDS.
// ---------------------------------------------------------------------------
__global__ __launch_bounds__(128) void fused_kernel(
    const float* __restrict__ pos_i, const float* __restrict__ pos_j,
    const float* __restrict__ v_j,
    const float* __restrict__ b1, const float* __restrict__ b2,
    const float* __restrict__ ws, float* __restrict__ out)
{
  __shared__ struct {
    float h2[EPB * HP];        // 67584 B  (stride 132: conflict-free b64 A-loads)
    float vv[EPB * VP];        // 18432 B  (v_aug, stride 36 -> 16B aligned rows)
    float chunk[CHUNK * 128];  // 67584 B  (Wf chunk = 4 j-blocks; h1 staging too)
  } sm;

  const int tid  = threadIdx.x;
  const int wave = tid >> 5;
  const int l    = tid & 31;
  const int lh   = l >> 4;   // lane half
  const int ln   = l & 15;

  const int eBlock = blockIdx.x * EPB;
  const int eWave  = eBlock + wave * EPW;

  const v2f* W1f = (const v2f*)(ws + W1F_OFF);
  const v2f* W2f = (const v2f*)(ws + W2F_OFF);
  const float* Wff = ws + WFF_OFF;

  float* h1w = &sm.chunk[wave * (16 * HP)];  // per-wave h1 tile (16 x 132)

  // ---- stage v_aug into LDS (one edge per lane, coalesced b128 reads) ----
  {
    const int le = wave * 32 + l;
    const v4f* vsrc = (const v4f*)(v_j + (size_t)(eBlock + le) * 32);
    v4f* vdst = (v4f*)&sm.vv[le * VP];
    #pragma unroll
    for (int q = 0; q < 8; ++q) vdst[q] = vsrc[q];
    sm.vv[le * VP + 32] = 1.0f;              // bias column of v_aug
  }

  // ---- phase 1+2: two 16-edge M-tiles per wave ----
  for (int m = 0; m < 2; ++m) {
    const int edge = eWave + m * 16 + ln;
    // layer-1 A fragment: K = {0,1}=pos_i.xy (lanes 0-15), {2,3}=pos_j.xy
    const float* psrc = (lh == 0) ? (pos_i + (size_t)edge * 2)
                                  : (pos_j + (size_t)edge * 2);
    v2f a1; a1.x = psrc[0]; a1.y = psrc[1];

    #pragma unroll
    for (int n = 0; n < 8; ++n) {
      v8f acc = {};
      v2f b = W1f[n * 32 + l];
      acc = __builtin_amdgcn_wmma_f32_16x16x4_f32(false, a1, false, b,
                                                  (short)0, acc, false, false);
      float bb = b1[n * 16 + ln];
      #pragma unroll
      for (int v = 0; v < 8; ++v) {
        float h = acc[v] + bb; h = h > 0.0f ? h : 0.0f;
        h1w[(v + 8 * lh) * HP + n * 16 + ln] = h;   // D-layout -> row-major
      }
    }

    // layer 2: hoist all 32 A fragments (b64 LDS reads, conflict-free)
    v2f a2[32];
    const v2f* h1v = (const v2f*)h1w;
    #pragma unroll
    for (int t = 0; t < 32; ++t)
      a2[t] = h1v[ln * (HP / 2) + 2 * t + lh];

    #pragma unroll 2
    for (int n = 0; n < 8; ++n) {
      v8f acc = {};
      #pragma unroll
      for (int t = 0; t < 32; ++t) {
        v2f b = W2f[(t * 8 + n) * 32 + l];
        acc = __builtin_amdgcn_wmma_f32_16x16x4_f32(false, a2[t], false, b,
                                                    (short)0, acc, false, false);
      }
      float bb = b2[n * 16 + ln];
      #pragma unroll
      for (int v = 0; v < 8; ++v) {
        float h = acc[v] + bb; h = h > 0.0f ? h : 0.0f;
        const int row = wave * 32 + m * 16 + v + 8 * lh;
        sm.h2[row * HP + n * 16 + ln] = h;
      }
    }
  }
  // h2 augmentation row: h=128 -> 1.0 (folds b3), h=129..131 -> 0
  {
    const int le = wave * 32 + l;
    v4f aug; aug.x = 1.0f; aug.y = 0.0f; aug.z = 0.0f; aug.w = 0.0f;
    *(v4f*)&sm.h2[le * HP + HID] = aug;
  }
  __syncthreads();   // all h1/h2 done before chunk buffer is recycled

  // ---- phase 3: out = T @ Wf, streaming Wf chunks through LDS ----
  // Chunk = 4 whole j-blocks -> branch-free inner loops of exactly 33 k-steps
  // with compile-time DS offsets.
  v8f acc[2][2] = {};                        // [m-tile][n-tile]
  const int le0 = wave * 32 + ln;            // M-tile 0 edge row (this lane)
  const int le1 = wave * 32 + 16 + ln;       // M-tile 1 edge row
  const v2f* a0base = (const v2f*)sm.h2 + le0 * (HP / 2) + lh;
  const v2f* a1base = (const v2f*)sm.h2 + le1 * (HP / 2) + lh;

  for (int c = 0; c < NCHUNK; ++c) {
    const int jb = (c < NCHUNK - 1) ? 4 : 1; // j-blocks in this chunk
    const int nsteps = jb * JBLK;
    {
      const v4f* src = (const v4f*)(Wff + (size_t)c * CHUNK * 128);
      v4f* dst = (v4f*)sm.chunk;
      const int nv = nsteps * 32;
      for (int i = tid; i < nv; i += 128) dst[i] = src[i];
    }
    if (c + 1 < NCHUNK) {                    // global_prefetch_b8 on next chunk
      const char* np = (const char*)(Wff + (size_t)(c + 1) * CHUNK * 128);
      __builtin_prefetch(np + (size_t)tid * 512, 0, 1);
    }
    __syncthreads();

    for (int jj = 0; jj < jb; ++jj) {
      const int j = c * 4 + jj;
      const float vj0 = sm.vv[le0 * VP + j];
      const float vj1 = sm.vv[le1 * VP + j];
      // re-base chunk pointer per j-block: inner offsets are immediates <= 16.5KB
      const v2f* cb = (const v2f*)sm.chunk + (size_t)jj * (JBLK * 64) + l;
      #pragma unroll
      for (int t = 0; t < JBLK; ++t) {       // h = 4*t, all offsets immediate
        v2f a0 = a0base[2 * t];
        v2f a1 = a1base[2 * t];
        v2f t0; t0.x = a0.x * vj0; t0.y = a0.y * vj0;   // T = h2 * v_aug[j]
        v2f t1; t1.x = a1.x * vj1; t1.y = a1.y * vj1;
        v2f b0 = cb[(t * 2 + 0) * 32];
        v2f b1f = cb[(t * 2 + 1) * 32];
        acc[0][0] = __builtin_amdgcn_wmma_f32_16x16x4_f32(false, t0, false, b0,
                                                          (short)0, acc[0][0], false, false);
        acc[0][1] = __builtin_amdgcn_wmma_f32_16x16x4_f32(false, t0, false, b1f,
                                                          (short)0, acc[0][1], false, false);
        acc[1][0] = __builtin_amdgcn_wmma_f32_16x16x4_f32(false, t1, false, b0,
                                                          (short)0, acc[1][0], false, false);
        acc[1][1] = __builtin_amdgcn_wmma_f32_16x16x4_f32(false, t1, false, b1f,
                                                          (short)0, acc[1][1], false, false);
      }
    }
    __syncthreads();
  }

  // ---- write out (D-layout: lane holds col N=ln, rows M=v+8*lh) ----
  #pragma unroll
  for (int m = 0; m < 2; ++m)
    #pragma unroll
    for (int n = 0; n < 2; ++n)
      #pragma unroll
      for (int v = 0; v < 8; ++v) {
        const int edge = eWave + m * 16 + v + 8 * lh;
        out[(size_t)edge * 32 + n * 16 + ln] = acc[m][n][v];
      }
}

// ---------------------------------------------------------------------------
extern "C" void kernel_launch(void* const* d_in, const int* in_sizes, int n_in,
                              void* d_out, int out_size, void* d_ws, size_t ws_size,
                              hipStream_t stream) {
  (void)n_in; (void)out_size; (void)ws_size;
  const float* pos_i = (const float*)d_in[0];
  const float* pos_j = (const float*)d_in[1];
  const float* v_j   = (const float*)d_in[2];
  const float* W1    = (const float*)d_in[3];
  const float* b1    = (const float*)d_in[4];
  const float* W2    = (const float*)d_in[5];
  const float* b2    = (const float*)d_in[6];
  const float* W3    = (const float*)d_in[7];
  const float* b3    = (const float*)d_in[8];
  float* out = (float*)d_out;
  float* ws  = (float*)d_ws;

  const int E = in_sizes[2] / 32;   // 262144

  prepack_kernel<<<(WS_FLOATS + 255) / 256, 256, 0, stream>>>(W1, W2, W3, b3, ws);
  fused_kernel<<<E / EPB, 128, 0, stream>>>(pos_i, pos_j, v_j, b1, b2, ws, out);
}